// InterpretCNN_88734024335655
// MI455X (gfx1250) — compile-verified
//
#include <hip/hip_runtime.h>
#include <math.h>

// ---------------------------------------------------------------------------
// InterpretCNN on MI455X (gfx1250, wave32).
// All GEMM stages use V_WMMA_F32_16X16X4_F32 (exact fp32, matrix pipe).
// fp32 WMMA VGPR layouts (ISA 7.12.2):
//   A 16x4 : lane L -> row M = L&15 ; VGPR0 = K = 2*(L>>4), VGPR1 = K+1
//   B 4x16 : lane L -> col N = L&15 ; VGPR0 = row K = 2*(L>>4), VGPR1 = K+1
//   C/D    : lane L -> col N = L&15 ; VGPR r -> row M = r + 8*(L>>4)
// Each wave owns a 16-row x (16*NTILES)-col output strip: one A fetch feeds
// NTILES WMMAs per k-step, amortizing vmem + address VALU over the matrix ops.
// ---------------------------------------------------------------------------

typedef __attribute__((ext_vector_type(2))) float v2f;
typedef __attribute__((ext_vector_type(8))) float v8f;

#define WMMA_F32(a, b, c) \
  __builtin_amdgcn_wmma_f32_16x16x4_f32(false, (a), false, (b), (short)0, (c), false, false)

#define BATCH  256
#define DSP    784     // H*W
#define CIN    64
#define NPROTO 100
#define NCLS   10
#define PPCK   10
#define ROWS   2560    // BATCH * PPC rows per class
#define F1     128
#define F2     64
#define F3     32
#define FD     1024
#define CONCAT 3200    // NPROTO * 32
#define CHUNK  320     // PPC * 32 (per-class concept width)

// PRUNE matrix packed as bitmask: bit k of kPruneMask[c] == PRUNE[c][k]
__constant__ unsigned short kPruneMask[NCLS] = {
    0x200, 0x012, 0x040, 0x200, 0x047, 0x116, 0x215, 0x086, 0x042, 0x060};

// ---------------------------------------------------------------------------
// Kernel 1: 1x1 conv + bias + ReLU, scattered into regrouped layout
//   x[c][b*10 + k][hw]  (class-major so stages 2-4 are dense GEMMs)
// GEMM: M = 200704 (b*784+hw), K = 64, N = 100 (padded to 128).
// Out-of-range columns: load index clamped only (no mask mul needed --
// WMMA columns are independent and clamped columns are never stored).
// ---------------------------------------------------------------------------
__global__ __launch_bounds__(128) void k_conv_regroup(
    const float* __restrict__ in, const float* __restrict__ cw,
    const float* __restrict__ cb, float* __restrict__ x) {
  const int lane = threadIdx.x & 31;
  const int wid  = blockIdx.x * (blockDim.x >> 5) + (threadIdx.x >> 5);
  const int mt = wid >> 2;        // 12544 m-tiles
  const int st = wid & 3;         // 4 strips of 32 cols
  const int m0 = mt << 4;
  const int n0 = st << 5;
  const int half = lane >> 4, lr = lane & 15;
  const float* Arow = in + (size_t)(m0 + lr) * CIN;

  int nc[2];
#pragma unroll
  for (int j = 0; j < 2; ++j) {
    const int nn = n0 + j * 16 + lr;
    nc[j] = (nn < NPROTO) ? nn : (NPROTO - 1);
  }

  v8f acc[2];
#pragma unroll
  for (int j = 0; j < 2; ++j) acc[j] = (v8f){0.f, 0.f, 0.f, 0.f, 0.f, 0.f, 0.f, 0.f};

#pragma unroll 4
  for (int k4 = 0; k4 < CIN; k4 += 4) {
    const int ka = k4 + (half << 1);
    v2f a = {Arow[ka], Arow[ka + 1]};
#pragma unroll
    for (int j = 0; j < 2; ++j) {
      v2f b = {cw[ka * NPROTO + nc[j]], cw[(ka + 1) * NPROTO + nc[j]]};
      acc[j] = WMMA_F32(a, b, acc[j]);
    }
  }

  // one uniform div per wave; a 16-row tile crosses at most one 784-boundary
  const int mb = m0 / DSP;
  const int mh = m0 % DSP;
#pragma unroll
  for (int j = 0; j < 2; ++j) {
    const int nn = n0 + j * 16 + lr;
    if (nn < NPROTO) {
      const int c = nn / PPCK, kk = nn % PPCK;
      const float bias = cb[nn];
#pragma unroll
      for (int r = 0; r < 8; ++r) {
        int hw = mh + r + (half << 3);
        int bb = mb;
        if (hw >= DSP) { hw -= DSP; bb += 1; }
        float v = acc[j][r] + bias;
        v = v > 0.f ? v : 0.f;
        x[((size_t)c * ROWS + bb * PPCK + kk) * DSP + hw] = v;
      }
    }
  }
}

// ---------------------------------------------------------------------------
// Kernels 2/3: per-class GEMM + bias + ReLU, compile-time shapes.
//   A: [NCLS][ROWS][KDIM], W: [NCLS][KDIM][NDIM], Y: [NCLS][ROWS][NDIM]
// NTILES accumulators per wave; one A fetch feeds NTILES WMMAs per k-step.
// ---------------------------------------------------------------------------
template <int KDIM, int NDIM, int NTILES>
__global__ __launch_bounds__(128) void k_class_gemm_relu(
    const float* __restrict__ A, const float* __restrict__ W,
    const float* __restrict__ bias, float* __restrict__ Y) {
  constexpr int NSTRIPS  = NDIM / (16 * NTILES);
  constexpr int PERCLASS = (ROWS / 16) * NSTRIPS;
  const int lane = threadIdx.x & 31;
  const int wid  = blockIdx.x * (blockDim.x >> 5) + (threadIdx.x >> 5);
  const int c   = wid / PERCLASS;
  const int rem = wid % PERCLASS;
  const int m0 = (rem / NSTRIPS) << 4;
  const int n0 = (rem % NSTRIPS) * (16 * NTILES);
  const int half = lane >> 4, lr = lane & 15;
  const float* Arow = A + ((size_t)c * ROWS + m0 + lr) * KDIM;
  const float* Wcol = W + (size_t)c * KDIM * NDIM + n0 + lr;

  v8f acc[NTILES];
#pragma unroll
  for (int j = 0; j < NTILES; ++j) acc[j] = (v8f){0.f, 0.f, 0.f, 0.f, 0.f, 0.f, 0.f, 0.f};

#pragma unroll 4
  for (int k4 = 0; k4 < KDIM; k4 += 4) {
    const int ka = k4 + (half << 1);
    v2f a = {Arow[ka], Arow[ka + 1]};
#pragma unroll
    for (int j = 0; j < NTILES; ++j) {
      v2f b = {Wcol[(size_t)ka * NDIM + j * 16],
               Wcol[(size_t)(ka + 1) * NDIM + j * 16]};
      acc[j] = WMMA_F32(a, b, acc[j]);
    }
  }

#pragma unroll
  for (int j = 0; j < NTILES; ++j) {
    const int nn = n0 + j * 16 + lr;
    const float bs = bias[c * NDIM + nn];
#pragma unroll
    for (int r = 0; r < 8; ++r) {
      const int m = m0 + r + (half << 3);
      float v = acc[j][r] + bs;
      v = v > 0.f ? v : 0.f;
      Y[((size_t)c * ROWS + m) * NDIM + nn] = v;
    }
  }
}

// ---------------------------------------------------------------------------
// Kernel 4: layer3 GEMM + tanh + norm (16-lane butterfly) + l2-normalize +
//           prune mask; writes proto / norm / weighted outputs.
// One wave owns a 16-row tile and both 16-wide N tiles (F3 = 32).
// ---------------------------------------------------------------------------
__global__ __launch_bounds__(128) void k_layer3_post(
    const float* __restrict__ h2, const float* __restrict__ w3,
    const float* __restrict__ b3, float* __restrict__ proto_o,
    float* __restrict__ norm_o, float* __restrict__ weighted_o) {
  const int lane = threadIdx.x & 31;
  const int wid  = blockIdx.x * (blockDim.x >> 5) + (threadIdx.x >> 5);
  const int c  = wid / (ROWS >> 4);
  const int m0 = (wid % (ROWS >> 4)) << 4;
  const int half = lane >> 4, lr = lane & 15;
  const float* Arow = h2 + ((size_t)c * ROWS + m0 + lr) * F2;
  const float* Wc   = w3 + (size_t)c * F2 * F3;

  v8f acc0 = {0.f, 0.f, 0.f, 0.f, 0.f, 0.f, 0.f, 0.f};
  v8f acc1 = {0.f, 0.f, 0.f, 0.f, 0.f, 0.f, 0.f, 0.f};
#pragma unroll 4
  for (int k4 = 0; k4 < F2; k4 += 4) {
    const int ka = k4 + (half << 1);
    v2f a  = {Arow[ka], Arow[ka + 1]};
    v2f b0 = {Wc[ka * F3 + lr],      Wc[(ka + 1) * F3 + lr]};
    v2f b1 = {Wc[ka * F3 + 16 + lr], Wc[(ka + 1) * F3 + 16 + lr]};
    acc0 = WMMA_F32(a, b0, acc0);
    acc1 = WMMA_F32(a, b1, acc1);
  }

  const float bs0 = b3[c * F3 + lr];
  const float bs1 = b3[c * F3 + 16 + lr];
  // one uniform div per wave; row = b*10 + k, 16-row tile wraps k at most twice
  const int mb = m0 / PPCK;
  const int mk = m0 % PPCK;
#pragma unroll
  for (int r = 0; r < 8; ++r) {
    int kk = mk + r + (half << 3);            // <= 9 + 15 = 24
    int bb = mb;
    if (kk >= 2 * PPCK) { kk -= 2 * PPCK; bb += 2; }
    else if (kk >= PPCK) { kk -= PPCK; bb += 1; }
    const float v0 = tanhf(acc0[r] + bs0);
    const float v1 = tanhf(acc1[r] + bs1);
    // sum of squares over all 32 features: reduce within each 16-lane half
    float part = v0 * v0 + v1 * v1;
    part += __shfl_xor(part, 1, 32);
    part += __shfl_xor(part, 2, 32);
    part += __shfl_xor(part, 4, 32);
    part += __shfl_xor(part, 8, 32);
    const float inv = rsqrtf(fmaxf(part, 1e-12f));
    const float wm  = ((kPruneMask[c] >> kk) & 1) ? 0.f : 1.f;
    const float p0 = v0 * inv, p1 = v1 * inv;
    const size_t base = ((size_t)(bb * NCLS + c) * PPCK + kk) * F3;
    proto_o[base + lr]         = p0;
    proto_o[base + 16 + lr]    = p1;
    weighted_o[base + lr]      = p0 * wm;
    weighted_o[base + 16 + lr] = p1 * wm;
    if (lr == 0)  // lane 0 covers rows m0+r, lane 16 covers rows m0+8+r
      norm_o[(size_t)(bb * NCLS + c) * PPCK + kk] = part;
  }
}

// ---------------------------------------------------------------------------
// Kernel 5: per-(b,c) relevance dot (320 elems) + sigmoid. One wave each.
// ---------------------------------------------------------------------------
__global__ __launch_bounds__(128) void k_relevance(
    const float* __restrict__ weighted_o, const float* __restrict__ rel_w,
    const float* __restrict__ rel_b, float* __restrict__ relev_o) {
  const int lane = threadIdx.x & 31;
  const int wid  = blockIdx.x * (blockDim.x >> 5) + (threadIdx.x >> 5);
  const int b = wid / NCLS, c = wid % NCLS;
  const float* wrow = weighted_o + (size_t)(b * NCLS + c) * CHUNK;
  const float* rw   = rel_w + (size_t)c * CHUNK;
  float s = 0.f;
#pragma unroll
  for (int j = lane; j < CHUNK; j += 32) s += wrow[j] * rw[j];
  s += __shfl_xor(s, 16, 32);
  s += __shfl_xor(s, 8, 32);
  s += __shfl_xor(s, 4, 32);
  s += __shfl_xor(s, 2, 32);
  s += __shfl_xor(s, 1, 32);
  if (lane == 0)
    relev_o[b * NCLS + c] = 1.f / (1.f + expf(-(s + rel_b[c])));
}

// ---------------------------------------------------------------------------
// Kernel 6: final dense. GEMM [256 x 3200] x [3200 x 1024] + bias + ReLU.
// Outer loop over the 10 classes hoists the per-class relevance scalar
// (A[b][d] = weighted[b][d] * relev[b][class]) -- no divides in the hot loop.
// 2-tile N-strips (16x32 per wave) -> 512 waves.
// ---------------------------------------------------------------------------
__global__ __launch_bounds__(128) void k_final_dense(
    const float* __restrict__ weighted_o, const float* __restrict__ relev_o,
    const float* __restrict__ fd_w, const float* __restrict__ fd_b,
    float* __restrict__ dense_o) {
  const int lane = threadIdx.x & 31;
  const int wid  = blockIdx.x * (blockDim.x >> 5) + (threadIdx.x >> 5);
  const int mt = wid >> 5, st = wid & 31;   // 32 strips of 32 cols (N = 1024)
  const int m0 = mt << 4, n0 = st << 5;
  const int half = lane >> 4, lr = lane & 15;
  const int row = m0 + lr;                  // batch index (always < 256)
  const float* wrow = weighted_o + (size_t)row * CONCAT;
  const float* rrow = relev_o + row * NCLS;
  const float* Wcol = fd_w + n0 + lr;

  v8f acc[2];
#pragma unroll
  for (int j = 0; j < 2; ++j) acc[j] = (v8f){0.f, 0.f, 0.f, 0.f, 0.f, 0.f, 0.f, 0.f};

  for (int cc = 0; cc < NCLS; ++cc) {
    const float rv = rrow[cc];
    const float* wch = wrow + cc * CHUNK;
    const float* Wch = Wcol + (size_t)cc * CHUNK * FD;
#pragma unroll 4
    for (int k4 = 0; k4 < CHUNK; k4 += 4) {
      const int ka = k4 + (half << 1);
      v2f a = {wch[ka] * rv, wch[ka + 1] * rv};
#pragma unroll
      for (int j = 0; j < 2; ++j) {
        v2f b = {Wch[(size_t)ka * FD + j * 16],
                 Wch[(size_t)(ka + 1) * FD + j * 16]};
        acc[j] = WMMA_F32(a, b, acc[j]);
      }
    }
  }

#pragma unroll
  for (int j = 0; j < 2; ++j) {
    const int nn = n0 + j * 16 + lr;
    const float bs = fd_b[nn];
#pragma unroll
    for (int r = 0; r < 8; ++r) {
      const int m = m0 + r + (half << 3);
      float v = acc[j][r] + bs;
      v = v > 0.f ? v : 0.f;
      dense_o[(size_t)m * FD + nn] = v;
    }
  }
}

// ---------------------------------------------------------------------------
extern "C" void kernel_launch(void* const* d_in, const int* in_sizes, int n_in,
                              void* d_out, int out_size, void* d_ws, size_t ws_size,
                              hipStream_t stream) {
  const float* inputs = (const float*)d_in[0];
  const float* conv_w = (const float*)d_in[1];
  const float* conv_b = (const float*)d_in[2];
  const float* w1     = (const float*)d_in[3];
  const float* b1     = (const float*)d_in[4];
  const float* w2     = (const float*)d_in[5];
  const float* b2     = (const float*)d_in[6];
  const float* w3     = (const float*)d_in[7];
  const float* b3     = (const float*)d_in[8];
  const float* rel_w  = (const float*)d_in[9];
  const float* rel_b  = (const float*)d_in[10];
  const float* fd_w   = (const float*)d_in[11];
  const float* fd_b   = (const float*)d_in[12];

  // d_out: proto | norm | relevances | dense_out | weighted (flat, fp32)
  float* out        = (float*)d_out;
  float* proto_o    = out;                    // 256*10*10*32 = 819200
  float* norm_o     = out + 819200;           // 25600
  float* relev_o    = out + 844800;           // 2560
  float* dense_o    = out + 847360;           // 262144
  float* weighted_o = out + 1109504;          // 819200

  // workspace: x (regrouped conv out), h1, h2   (~100 MB, L2-resident reuse)
  float* ws = (float*)d_ws;
  float* x  = ws;                             // 10*2560*784
  float* h1 = x + (size_t)NCLS * ROWS * DSP;  // 10*2560*128
  float* h2 = h1 + (size_t)NCLS * ROWS * F1;  // 10*2560*64

  const dim3 blk(128);  // 4 waves/block; all grids exact -> EXEC always full
  // kernel1: 12544 m-tiles * 4 strips = 50176 waves -> 12544 blocks
  k_conv_regroup<<<12544, blk, 0, stream>>>(inputs, conv_w, conv_b, x);
  // layer1: 10 * 160 m-tiles * 2 strips = 3200 waves -> 800 blocks
  k_class_gemm_relu<DSP, F1, 4><<<800, blk, 0, stream>>>(x, w1, b1, h1);
  // layer2: 10 * 160 * 1 strip = 1600 waves -> 400 blocks
  k_class_gemm_relu<F1, F2, 4><<<400, blk, 0, stream>>>(h1, w2, b2, h2);
  // layer3 + post: 10 * 160 = 1600 waves -> 400 blocks
  k_layer3_post<<<400, blk, 0, stream>>>(h2, w3, b3, proto_o, norm_o, weighted_o);
  // relevance: 2560 waves -> 640 blocks
  k_relevance<<<640, blk, 0, stream>>>(weighted_o, rel_w, rel_b, relev_o);
  // final dense: 16 m-tiles * 32 strips = 512 waves -> 128 blocks
  k_final_dense<<<128, blk, 0, stream>>>(weighted_o, relev_o, fd_w, fd_b, dense_o);
}